// custom_loss_21114059227388
// MI455X (gfx1250) — compile-verified
//
#include <hip/hip_runtime.h>
#include <stdint.h>

// Problem constants (match setup_inputs: B=64, C=1, H=W=512, fp32)
#define IMG_H   512
#define IMG_W   512
#define NBATCH  64
#define BAND    32                 // output rows per workgroup
#define INROWS  (BAND + 2)         // input rows staged in LDS (with halo)
#define NBANDS  (IMG_H / BAND)     // 16
#define CROP    5
#define NTHREADS 512

// Separable factorization of the 3x3 window: w = [p,q,p] (x) [p,q,p]
// p^2 = 0.09474166, q^2 = 0.14776132, p*q = 0.11831801 (Gaussian, sum = 1)
#define FP 0.30780131f
#define FQ 0.38439733f

struct H5 { float x, y, xx, yy, xy; };

__global__ __launch_bounds__(NTHREADS, 1)
void ssim_band_kernel(const float* __restrict__ X,
                      const float* __restrict__ Y,
                      float* __restrict__ partial) {
  // 2 x 34 x 512 floats = 136 KiB staged in LDS (320 KiB available per WGP)
  __shared__ __align__(16) float ldsX[INROWS * IMG_W];
  __shared__ __align__(16) float ldsY[INROWS * IMG_W];
  __shared__ float red[NTHREADS / 32];

  const int tid  = threadIdx.x;
  const int img  = blockIdx.x >> 4;            // blockIdx.x / NBANDS
  const int band = blockIdx.x & (NBANDS - 1);
  const int r0   = band * BAND;

  // Input rows to stage: [r0-1, r0+32] clamped to the image. Slot s in LDS
  // always corresponds to input row (r0 - 1 + s); band 0 leaves slot 0
  // unloaded (no valid output row ever reads it, since output rows < 5 are
  // cropped away).
  const int rlo   = (r0 == 0) ? 0 : (r0 - 1);
  const int rhi   = (r0 + BAND < IMG_H) ? (r0 + BAND) : (IMG_H - 1);
  const int nrows = rhi - rlo + 1;             // 33 or 34
  const int slot0 = rlo - (r0 - 1);            // 1 for band 0, else 0

  // The staged band is fully contiguous in global memory: flat async DMA.
  const uint64_t gbyte = ((uint64_t)img * IMG_H * IMG_W + (uint64_t)rlo * IMG_W) * 4ull;
  const uint64_t xbase = (uint64_t)(uintptr_t)X + gbyte;
  const uint64_t ybase = (uint64_t)(uintptr_t)Y + gbyte;

  // Low 32 bits of a generic pointer into LDS == LDS byte address (aperture
  // scheme, ISA ch.10.2); rows are 2048 B so 16 B alignment holds everywhere.
  const uint32_t ldsXb = (uint32_t)(uintptr_t)(&ldsX[slot0 * IMG_W]);
  const uint32_t ldsYb = (uint32_t)(uintptr_t)(&ldsY[slot0 * IMG_W]);

  // Async global -> LDS DMA, 16 B per lane per issue (ASYNCcnt-tracked).
  const int chunks = nrows * (IMG_W / 4);      // 16-byte chunks per image band
  for (int j = tid; j < chunks; j += NTHREADS) {
    const uint32_t off = (uint32_t)j * 16u;
    asm volatile("global_load_async_to_lds_b128 %0, %1, %2"
                 :: "v"(ldsXb + off), "v"(off), "s"(xbase) : "memory");
    asm volatile("global_load_async_to_lds_b128 %0, %1, %2"
                 :: "v"(ldsYb + off), "v"(off), "s"(ybase) : "memory");
  }
  asm volatile("s_wait_asynccnt 0x0" ::: "memory");
  __syncthreads();

  // Each thread owns one column; horizontal 3-tap once per input row,
  // vertical 3-tap via register rotation.
  const int c  = tid;
  const int cl = (c == 0) ? 0 : (c - 1);
  const int cr = (c == IMG_W - 1) ? (IMG_W - 1) : (c + 1);

  auto hrow = [&](int s) -> H5 {
    const int base = s * IMG_W;
    const float xl = ldsX[base + cl], xc = ldsX[base + c], xr = ldsX[base + cr];
    const float yl = ldsY[base + cl], yc = ldsY[base + c], yr = ldsY[base + cr];
    H5 h;
    h.x  = FP * (xl + xr) + FQ * xc;
    h.y  = FP * (yl + yr) + FQ * yc;
    h.xx = FP * fmaf(xl, xl, xr * xr) + FQ * (xc * xc);
    h.yy = FP * fmaf(yl, yl, yr * yr) + FQ * (yc * yc);
    h.xy = FP * fmaf(xl, yl, xr * yr) + FQ * (xc * yc);
    return h;
  };

  const int rout_lo = (r0 > CROP) ? r0 : CROP;
  const int rout_hi_band = r0 + BAND - 1;
  const int rout_hi = (rout_hi_band < (IMG_H - 1 - CROP)) ? rout_hi_band
                                                          : (IMG_H - 1 - CROP);

  float acc = 0.0f;
  H5 h0 = hrow(rout_lo - r0);       // input row rout_lo - 1
  H5 h1 = hrow(rout_lo - r0 + 1);   // input row rout_lo
  for (int r = rout_lo; r <= rout_hi; ++r) {
    H5 h2 = hrow(r - r0 + 2);       // input row r + 1
    const float vx  = FP * (h0.x  + h2.x ) + FQ * h1.x;   // mu1
    const float vy  = FP * (h0.y  + h2.y ) + FQ * h1.y;   // mu2
    const float vxx = FP * (h0.xx + h2.xx) + FQ * h1.xx;  // E[x^2]
    const float vyy = FP * (h0.yy + h2.yy) + FQ * h1.yy;  // E[y^2]
    const float vxy = FP * (h0.xy + h2.xy) + FQ * h1.xy;  // E[xy]
    const float s1  = vxx - vx * vx;
    const float s2  = vyy - vy * vy;
    const float s12 = vxy - vx * vy;
    acc += fmaf(s1, s2, -2.0f * s12);
    h0 = h1; h1 = h2;
  }
  // Column crop: discard border columns.
  if (c < CROP || c > IMG_W - 1 - CROP) acc = 0.0f;

  // wave32 reduction, then cross-wave via LDS.
  for (int off = 16; off > 0; off >>= 1)
    acc += __shfl_xor(acc, off, 32);
  const int wid  = tid >> 5;
  const int lane = tid & 31;
  if (lane == 0) red[wid] = acc;
  __syncthreads();
  if (tid < 16) {
    float v = red[tid];
    v += __shfl_xor(v, 8, 32);
    v += __shfl_xor(v, 4, 32);
    v += __shfl_xor(v, 2, 32);
    v += __shfl_xor(v, 1, 32);
    if (tid == 0) {
      const float inv_area = 1.0f / (502.0f * 502.0f);
      partial[blockIdx.x] = v * inv_area;   // per-sample mean contribution
    }
  }
}

// Deterministic final reduction over the 1024 per-block partials.
__global__ __launch_bounds__(NTHREADS, 1)
void loss_reduce_kernel(const float* __restrict__ partial,
                        float* __restrict__ out) {
  __shared__ float red[NTHREADS / 32];
  const int tid = threadIdx.x;
  float v = partial[tid] + partial[tid + NTHREADS];   // 1024 partials
  for (int off = 16; off > 0; off >>= 1)
    v += __shfl_xor(v, off, 32);
  if ((tid & 31) == 0) red[tid >> 5] = v;
  __syncthreads();
  if (tid < 16) {
    float s = red[tid];
    s += __shfl_xor(s, 8, 32);
    s += __shfl_xor(s, 4, 32);
    s += __shfl_xor(s, 2, 32);
    s += __shfl_xor(s, 1, 32);
    if (tid == 0) out[0] = s;
  }
}

extern "C" void kernel_launch(void* const* d_in, const int* in_sizes, int n_in,
                              void* d_out, int out_size, void* d_ws, size_t ws_size,
                              hipStream_t stream) {
  (void)in_sizes; (void)n_in; (void)out_size; (void)ws_size;
  const float* X = (const float*)d_in[0];
  const float* Y = (const float*)d_in[1];
  float* part = (float*)d_ws;                  // 1024 floats of scratch
  float* out  = (float*)d_out;

  ssim_band_kernel<<<dim3(NBATCH * NBANDS), dim3(NTHREADS), 0, stream>>>(X, Y, part);
  loss_reduce_kernel<<<dim3(1), dim3(NTHREADS), 0, stream>>>(part, out);
}